// RTDetrv2MultiscaleDeformableAttention_51479478009968
// MI455X (gfx1250) — compile-verified
//
#include <hip/hip_runtime.h>
#include <math.h>

// ---------------- problem constants ----------------
#define B_SZ     16
#define Q_SZ     300
#define S_SZ     8400      // 80*80 + 40*40 + 20*20
#define DM       256
#define NHEADS   8
#define HDIM     32
#define LPN      12        // N_LEVELS * N_POINTS

typedef float v2f __attribute__((ext_vector_type(2)));
typedef float v8f __attribute__((ext_vector_type(8)));
typedef int   v4i __attribute__((ext_vector_type(4)));

typedef __attribute__((address_space(1))) v4i* gptr_v4i;
typedef __attribute__((address_space(3))) v4i* lptr_v4i;

__device__ __forceinline__ int lvl_w(int l)    { return 80 >> l; }
__device__ __forceinline__ int lvl_h(int l)    { return 80 >> l; }
__device__ __forceinline__ int lvl_base(int l) { return l == 0 ? 0 : (l == 1 ? 6400 : 8000); }

#if __has_builtin(__builtin_amdgcn_global_load_async_to_lds_b128)
#define HAVE_ASYNC_LDS 1
#else
#define HAVE_ASYNC_LDS 0
#endif

// =====================================================================
// WMMA f32 GEMM with LDS-staged weights:
//   C[M,N] = A[M,K] @ W[N,K]^T + bias[N]
// Workgroup = 8 waves; W (N x K) staged once into LDS via async
// global->LDS b128 (row stride K+4 floats: rows stay 16B-aligned and
// the 16 B-fragment lanes land in distinct bank groups). Each wave
// computes one 16-row M-tile across the full N (NTILES accumulators),
// amortizing the staging over 128 A-rows per workgroup.
// Inner loop is double-buffered on BOTH operands: the A fragment
// (global b64, non-temporal) and all NTILES B fragments (ds_load_b64)
// for step k+4 are issued before the WMMAs for step k, so the matrix
// pipe never waits on a just-issued load.
// Fragment layout (ISA 7.12.2, 32-bit, wave32): lane l, lh=l&15, sel=l>>4
//   A frag = A[m0+lh][k+2*sel .. +1]
//   B frag = W[col][k+2*sel .. +1]
//   C/D  : VGPR r -> C[m0 + 8*sel + r][col]
// =====================================================================
template<int NTILES>
__global__ __launch_bounds__(256)
void wmma_gemm_lds(const float* __restrict__ A,
                   const float* __restrict__ W,
                   const float* __restrict__ bias,
                   float* __restrict__ C,
                   int M, int N, int K) {
  extern __shared__ float ldsW[];          // N rows x (K+4) floats
  const int tid  = threadIdx.x;
  const int lane = tid & 31;
  const int wid  = tid >> 5;
  const int KP   = K + 4;

  // ---- stage W into LDS (async global->LDS on CDNA5) ----
  {
    const int chunks = K / 128;            // 32 lanes * 4 floats per chunk
    for (int row = wid; row < N; row += 8) {
      const float* grow = W + (size_t)row * K;
      float*       lrow = ldsW + (size_t)row * KP;
#pragma unroll
      for (int c = 0; c < 4; ++c) {
        if (c >= chunks) break;
        const int fo = c * 128 + lane * 4;
#if HAVE_ASYNC_LDS
        __builtin_amdgcn_global_load_async_to_lds_b128(
            (gptr_v4i)(grow + fo),
            (lptr_v4i)(lrow + fo),
            /*imm_offset=*/0, /*cpol=*/0);
#else
        *(float4*)(lrow + fo) = *(const float4*)(grow + fo);
#endif
      }
    }
#if HAVE_ASYNC_LDS
    asm volatile("s_wait_asynccnt 0x0" ::: "memory");
#endif
    __syncthreads();
  }

  const int m0 = (blockIdx.x * 8 + wid) * 16;
  if (m0 < M) {                            // wave-uniform: EXEC all-ones for WMMA
    const int lh  = lane & 15;
    const int sel = lane >> 4;

    v8f acc[NTILES];
#pragma unroll
    for (int t = 0; t < NTILES; ++t) acc[t] = (v8f)0.0f;

    const float* arow = A + (size_t)(m0 + lh) * K + 2 * sel;
    const float* brow = ldsW + lh * KP + 2 * sel;   // + t*16*KP + k

    // prologue: fragments for k = 0
    v2f af = __builtin_nontemporal_load((const v2f*)arow);
    v2f bf[NTILES];
#pragma unroll
    for (int t = 0; t < NTILES; ++t)
      bf[t] = *(const v2f*)(brow + (size_t)(16 * t) * KP);

#pragma unroll 2
    for (int k = 0; k < K; k += 4) {
      const int kn = (k + 4 < K) ? (k + 4) : k;
      // issue next-step loads first (independent of this step's WMMAs)
      v2f an = __builtin_nontemporal_load((const v2f*)(arow + kn));
      v2f bn[NTILES];
#pragma unroll
      for (int t = 0; t < NTILES; ++t)
        bn[t] = *(const v2f*)(brow + (size_t)(16 * t) * KP + kn);
      // compute with current fragments
#pragma unroll
      for (int t = 0; t < NTILES; ++t)
        acc[t] = __builtin_amdgcn_wmma_f32_16x16x4_f32(
            false, af, false, bf[t], (short)0, acc[t], false, false);
      af = an;
#pragma unroll
      for (int t = 0; t < NTILES; ++t) bf[t] = bn[t];
    }

#pragma unroll
    for (int t = 0; t < NTILES; ++t) {
      const int col = 16 * t + lh;
      const float bv = bias ? bias[col] : 0.0f;
#pragma unroll
      for (int r = 0; r < 8; ++r) {
        const int row = m0 + 8 * sel + r;
        C[(size_t)row * N + col] = acc[t][r] + bv;
      }
    }
  }
}

// =====================================================================
// Per-(b,q,h): softmax over 12 logits, absolute pixel sampling coords:
//   g = (ref_xy + off * 0.25 * ref_wh * 0.5) * dim - 0.5
// =====================================================================
__global__ __launch_bounds__(256)
void loc_softmax_kernel(const float* __restrict__ off_raw,   // [B*Q, H*LP*2]
                        const float* __restrict__ attn_raw,  // [B*Q, H*LP]
                        const float* __restrict__ refp,      // [B*Q, 4]
                        float2* __restrict__ locs,           // [B*Q*H, LP]
                        float*  __restrict__ attw) {         // [B*Q*H, LP]
  const int t = blockIdx.x * blockDim.x + threadIdx.x;
  if (t >= B_SZ * Q_SZ * NHEADS) return;
  const int h  = t % NHEADS;
  const int bq = t / NHEADS;

  const float* r = refp + (size_t)bq * 4;
  const float cx = r[0], cy = r[1];
  const float rw = r[2] * (0.25f * 0.5f);
  const float rh = r[3] * (0.25f * 0.5f);

  const float* al = attn_raw + (size_t)bq * (NHEADS * LPN) + h * LPN;
  float lg[LPN], m = al[0];
#pragma unroll
  for (int i = 0; i < LPN; ++i) { lg[i] = al[i]; m = fmaxf(m, lg[i]); }
  float sum = 0.0f;
#pragma unroll
  for (int i = 0; i < LPN; ++i) { lg[i] = __expf(lg[i] - m); sum += lg[i]; }
  const float inv = 1.0f / sum;

  const float* ol = off_raw + (size_t)bq * (NHEADS * LPN * 2) + h * LPN * 2;
#pragma unroll
  for (int i = 0; i < LPN; ++i) {
    const int lvl = i >> 2;
    const float lx = cx + ol[2 * i + 0] * rw;
    const float ly = cy + ol[2 * i + 1] * rh;
    const size_t o = (size_t)t * LPN + i;
    locs[o] = make_float2(lx * (float)lvl_w(lvl) - 0.5f,
                          ly * (float)lvl_h(lvl) - 0.5f);
    attw[o] = lg[i] * inv;
  }
}

// =====================================================================
// Bilinear sampling + attention-weighted sum.
// One wave32 per (b,q,h); lane = head-dim channel -> every tap is a
// fully coalesced 32-lane contiguous load; tap coords are wave-uniform.
// =====================================================================
__global__ __launch_bounds__(256)
void msda_sample_kernel(const float* __restrict__ value,   // [B, S, 256]
                        const float2* __restrict__ locs,   // [B*Q*H, LP]
                        const float*  __restrict__ attw,   // [B*Q*H, LP]
                        float* __restrict__ out_heads) {   // [B*Q, 256]
  const int wave = blockIdx.x * (blockDim.x >> 5) + (threadIdx.x >> 5);
  const int lane = threadIdx.x & 31;
  if (wave >= B_SZ * Q_SZ * NHEADS) return;
  const int h  = wave % NHEADS;
  const int bq = wave / NHEADS;
  const int b  = bq / Q_SZ;

  const float* vb = value + (size_t)b * S_SZ * DM + h * HDIM + lane;
  float acc = 0.0f;

#pragma unroll
  for (int i = 0; i < LPN; ++i) {
    const int lvl = i >> 2;
    const int Wd = lvl_w(lvl), Hd = lvl_h(lvl), base = lvl_base(lvl);
    const float2 g = locs[(size_t)wave * LPN + i];
    const float aw = attw[(size_t)wave * LPN + i];

    const float x0f = floorf(g.x), y0f = floorf(g.y);
    const float fx = g.x - x0f, fy = g.y - y0f;
    const int ix0 = (int)x0f, iy0 = (int)y0f;
    const float tw[4] = {(1.0f - fx) * (1.0f - fy), fx * (1.0f - fy),
                         (1.0f - fx) * fy,          fx * fy};

#pragma unroll
    for (int tap = 0; tap < 4; ++tap) {
      const int ix = ix0 + (tap & 1);
      const int iy = iy0 + (tap >> 1);
      if (ix >= 0 && ix < Wd && iy >= 0 && iy < Hd) {
        const float v = vb[(size_t)(base + iy * Wd + ix) * DM];
        acc += aw * tw[tap] * v;
      }
    }
  }
  out_heads[(size_t)wave * HDIM + lane] = acc;   // == [bq*256 + h*32 + lane]
}

// =====================================================================
// Launch
// =====================================================================
extern "C" void kernel_launch(void* const* d_in, const int* in_sizes, int n_in,
                              void* d_out, int out_size, void* d_ws, size_t ws_size,
                              hipStream_t stream) {
  (void)in_sizes; (void)n_in; (void)out_size; (void)ws_size;

  const float* hidden  = (const float*)d_in[0];   // [16,300,256]
  const float* enc     = (const float*)d_in[1];   // [16,8400,256]
  const float* refp    = (const float*)d_in[2];   // [16,300,1,4]
  const float* w_value = (const float*)d_in[3];   // [256,256]
  const float* b_value = (const float*)d_in[4];
  const float* w_off   = (const float*)d_in[5];   // [192,256]
  const float* b_off   = (const float*)d_in[6];
  const float* w_attn  = (const float*)d_in[7];   // [96,256]
  const float* b_attn  = (const float*)d_in[8];
  const float* w_out   = (const float*)d_in[9];   // [256,256]
  const float* b_out   = (const float*)d_in[10];

  // ------ workspace carve-out ------
  char* ws = (char*)d_ws;
  float*  value    = (float*)ws;  ws += (size_t)B_SZ * S_SZ * DM * 4;           // 137.6 MB
  float*  off_raw  = (float*)ws;  ws += (size_t)B_SZ * Q_SZ * NHEADS*LPN*2 * 4;
  float*  attn_raw = (float*)ws;  ws += (size_t)B_SZ * Q_SZ * NHEADS*LPN * 4;
  float2* locs     = (float2*)ws; ws += (size_t)B_SZ * Q_SZ * NHEADS*LPN * 8;
  float*  attw     = (float*)ws;  ws += (size_t)B_SZ * Q_SZ * NHEADS*LPN * 4;
  float*  out_hd   = (float*)ws;

  const int MV = B_SZ * S_SZ;   // 134400
  const int MQ = B_SZ * Q_SZ;   // 4800
  const int K  = DM;            // 256

  auto lds_bytes = [](int N, int K_) { return (size_t)N * (K_ + 4) * sizeof(float); };

  // 1. value = enc @ w_value^T + b_value        (16 n-tiles per wave)
  wmma_gemm_lds<16><<<(MV / 16 + 7) / 8, 256, lds_bytes(256, K), stream>>>(
      enc, w_value, b_value, value, MV, 256, K);

  // 2. off_raw = hidden @ w_offsets^T + b_offsets (12 n-tiles)
  wmma_gemm_lds<12><<<(MQ / 16 + 7) / 8, 256, lds_bytes(192, K), stream>>>(
      hidden, w_off, b_off, off_raw, MQ, 192, K);

  // 3. attn_raw = hidden @ w_attn^T + b_attn     (6 n-tiles)
  wmma_gemm_lds<6><<<(MQ / 16 + 7) / 8, 256, lds_bytes(96, K), stream>>>(
      hidden, w_attn, b_attn, attn_raw, MQ, 96, K);

  // 4. softmax + sampling locations
  {
    int threads = MQ * NHEADS;  // 38400
    loc_softmax_kernel<<<(threads + 255) / 256, 256, 0, stream>>>(off_raw, attn_raw,
                                                                  refp, locs, attw);
  }
  // 5. bilinear sampling + weighted sum (wave per (b,q,h))
  {
    int waves = MQ * NHEADS;    // 38400
    msda_sample_kernel<<<(waves + 7) / 8, 256, 0, stream>>>(value, locs, attw, out_hd);
  }
  // 6. d_out = out_hd @ w_out^T + b_out
  wmma_gemm_lds<16><<<(MQ / 16 + 7) / 8, 256, lds_bytes(256, K), stream>>>(
      out_hd, w_out, b_out, (float*)d_out, MQ, 256, K);
}